// LightGCN_7146825581233
// MI455X (gfx1250) — compile-verified
//
#include <hip/hip_runtime.h>

// LightGCN propagation for MI455X (gfx1250, wave32).
// Memory-bound scatter/gather: optimize HBM traffic; use WMMA f32 16x16x4
// (identity-B trick) for the fused dense accumulate (+ final scale) passes.

#define N_USER   300000
#define N_ITEM   200000
#define N_NODES  500000
#define DIM      64
#define N_EDGES  1250000
#define N_LAYERS 3

typedef float v2f __attribute__((ext_vector_type(2)));
typedef float v8f __attribute__((ext_vector_type(8)));

// ---------------------------------------------------------------------------
// h0 = concat(user_emb, item_emb).  float4-vectorized.
// ---------------------------------------------------------------------------
__global__ void lg_init(const float4* __restrict__ user,
                        const float4* __restrict__ item,
                        float4* __restrict__ h) {
    size_t i = (size_t)blockIdx.x * blockDim.x + threadIdx.x;
    const size_t nUser4 = (size_t)N_USER * DIM / 4;
    const size_t nTot4  = (size_t)N_NODES * DIM / 4;
    if (i >= nTot4) return;
    h[i] = (i < nUser4) ? user[i] : item[i - nUser4];
}

__global__ void lg_zero(float4* __restrict__ p, int n4) {
    size_t i = (size_t)blockIdx.x * blockDim.x + threadIdx.x;
    if (i < (size_t)n4) p[i] = make_float4(0.f, 0.f, 0.f, 0.f);
}

// ---------------------------------------------------------------------------
// Weighted scatter: hn[dst] += w * h[src].
// 16 lanes per edge (one float4 each) -> 2 edges per wave32; coalesced 256B
// row gathers; f32 atomics resolve in L2 (128MB dest ~fits in 192MB L2).
// global_prefetch_b8 on a future edge's source row hides gather latency.
// ---------------------------------------------------------------------------
__global__ void lg_scatter(const float* __restrict__ h,
                           const float* __restrict__ w,
                           const int*   __restrict__ src,
                           const int*   __restrict__ dst,
                           float*       __restrict__ hn) {
    size_t tid = (size_t)blockIdx.x * blockDim.x + threadIdx.x;
    int e = (int)(tid >> 4);
    if (e >= N_EDGES) return;
    int q = (int)(tid & 15);           // lane's float4 slot within the row

    int   s  = src[e];
    int   d  = dst[e];
    float wv = w[e];

    // Prefetch the exact addresses this lane slot will gather ~128 edges
    // from now (emits gfx1250 global_prefetch_b8; speculative, RT).
    int ea = e + 128;
    if (ea < N_EDGES) {
        int sa = src[ea];
        __builtin_prefetch(h + (size_t)sa * DIM + q * 4, 0, 1);
    }

    const float4 v = *(const float4*)(h + (size_t)s * DIM + q * 4);
    float* out = hn + (size_t)d * DIM + q * 4;
    atomicAdd(out + 0, v.x * wv);
    atomicAdd(out + 1, v.y * wv);
    atomicAdd(out + 2, v.z * wv);
    atomicAdd(out + 3, v.w * wv);
}

// ---------------------------------------------------------------------------
// Fused dense pass on 16x16 f32 tiles via V_WMMA_F32_16X16X4_F32:
//   Om_tile = scale * Tm_tile + scale * Cm_tile
// C is pre-scaled at load (8 v_mul_f32); the T term goes through the matrix
// pipe as D = sum_{kc=0..3} A_kc x B_kc + C with B_kc = scale * (selection of
// columns 4kc..4kc+3 of I16), built in registers.
//
// 32-bit layouts (ISA 7.12.2), lane l = 16*hh + q, hh in {0,1}, q in [0,16):
//   A (16x4):  a[j] = T[q][4kc + 2*hh + j],  j in {0,1}
//   B (4x16):  b[j] = B[2*hh + j][q]
//   C/D:       c[v] = T[(v + 8*hh)][q],      v in [0,8)
// One wave per tile; grid divides exactly -> EXEC all-ones (WMMA requirement).
// ---------------------------------------------------------------------------
__global__ void lg_wmma_axpy(const float* __restrict__ Tm,
                             const float* __restrict__ Cm,
                             float*       __restrict__ Om,
                             float scale) {
    int lane = threadIdx.x & 31;
    int wid  = threadIdx.x >> 5;
    int tile = blockIdx.x * (blockDim.x >> 5) + wid;
    const int nTiles = (N_NODES / 16) * (DIM / 16);   // 125000
    if (tile >= nTiles) return;                        // wave-uniform exit

    int rowBlk = tile >> 2;        // 4 column-tiles per 16-row block (DIM=64)
    int colT   = tile & 3;
    size_t base = (size_t)rowBlk * 16 * DIM + (size_t)colT * 16;

    int hh = lane >> 4;
    int q  = lane & 15;

    const float* Tp = Tm + base;
    const float* Cp = Cm + base;
    float*       Op = Om + base;

    v8f c;
#pragma unroll
    for (int v = 0; v < 8; ++v)
        c[v] = scale * Cp[(size_t)(v + 8 * hh) * DIM + q];   // pre-scaled C

#pragma unroll
    for (int kc = 0; kc < 4; ++kc) {
        v2f a, b;
        const float* ap = Tp + (size_t)q * DIM + 4 * kc + 2 * hh;
        a.x = ap[0];
        a.y = ap[1];
        int c0 = 4 * kc + 2 * hh;                        // B = scale * I-select
        b.x = (q == c0)     ? scale : 0.0f;
        b.y = (q == c0 + 1) ? scale : 0.0f;
        // (neg_a, A, neg_b, B, c_mod, C, reuse_a, reuse_b)
        c = __builtin_amdgcn_wmma_f32_16x16x4_f32(false, a, false, b,
                                                  (short)0, c, false, false);
    }

#pragma unroll
    for (int v = 0; v < 8; ++v)
        Op[(size_t)(v + 8 * hh) * DIM + q] = c[v];
}

// ---------------------------------------------------------------------------
extern "C" void kernel_launch(void* const* d_in, const int* in_sizes, int n_in,
                              void* d_out, int out_size, void* d_ws, size_t ws_size,
                              hipStream_t stream) {
    const float* user = (const float*)d_in[0];
    const float* item = (const float*)d_in[1];
    const float* ew   = (const float*)d_in[2];
    const int*   src  = (const int*)d_in[3];
    const int*   dst  = (const int*)d_in[4];

    float* acc  = (float*)d_out;                         // accumulator = output
    float* bufA = (float*)d_ws;                          // hop buffer A (128MB)
    float* bufB = bufA + (size_t)N_NODES * DIM;          // hop buffer B (128MB)

    const int n4 = N_NODES * DIM / 4;                    // 8,000,000 float4
    dim3 blk(256);                                       // 8 waves / block

    const int nInitBlk    = (n4 + 255) / 256;            // 31250
    const int nScatterBlk = (N_EDGES * 16 + 255) / 256;  // 78125
    const int nTileBlk    = ((N_NODES / 16) * (DIM / 16)) / 8; // 15625 exact

    // h0 = concat(user, item)
    lg_init<<<nInitBlk, blk, 0, stream>>>((const float4*)user,
                                          (const float4*)item, (float4*)bufA);

    // Layer 1: h1 = S(h0);  acc = h1 + h0
    lg_zero<<<nInitBlk, blk, 0, stream>>>((float4*)bufB, n4);
    lg_scatter<<<nScatterBlk, blk, 0, stream>>>(bufA, ew, src, dst, bufB);
    lg_wmma_axpy<<<nTileBlk, blk, 0, stream>>>(bufB, bufA, acc, 1.0f);

    // Layer 2: h2 = S(h1);  acc = h2 + acc
    lg_zero<<<nInitBlk, blk, 0, stream>>>((float4*)bufA, n4);
    lg_scatter<<<nScatterBlk, blk, 0, stream>>>(bufB, ew, src, dst, bufA);
    lg_wmma_axpy<<<nTileBlk, blk, 0, stream>>>(bufA, acc, acc, 1.0f);

    // Layer 3 (fused mean): h3 = S(h2);  out = 0.25*h3 + 0.25*acc
    lg_zero<<<nInitBlk, blk, 0, stream>>>((float4*)bufB, n4);
    lg_scatter<<<nScatterBlk, blk, 0, stream>>>(bufA, ew, src, dst, bufB);
    lg_wmma_axpy<<<nTileBlk, blk, 0, stream>>>(bufB, acc, acc, 0.25f);
}